// G2C_111669150124
// MI455X (gfx1250) — compile-verified
//
#include <hip/hip_runtime.h>
#include <hip/hip_bf16.h>

typedef __attribute__((ext_vector_type(2))) float v2f;
typedef __attribute__((ext_vector_type(8))) float v8f;

// ---------------------------------------------------------------------------
// Problem dims (fixed by the reference)
// ---------------------------------------------------------------------------
#define NSIDE 512                 // N = H = 512
#define NROWS (512 * 512)         // rows of gnn_edge_out
#define T_STEPS 100

// ---------------------------------------------------------------------------
// Kernel 1: fused  h = relu(X@W1 + b1);  emb += sum(h);  e = h@W2 + b2
//           writes D_init = softplus(e0)*(1-eye), W = softplus(e1)
// WMMA f32 16x16x4, M-block = 64 rows, full N = 512 per workgroup.
// ---------------------------------------------------------------------------
constexpr int MB = 64;   // rows per workgroup
constexpr int KC = 16;   // K chunk staged in LDS
constexpr int SA = 20;   // ldsA row stride (floats)   64*20  = 1280
constexpr int SB = 20;   // ldsB col stride (floats)   512*20 = 10240
constexpr int SH = 65;   // ldsH row stride (floats)   64*65  = 4160 (aliases A/B)

__global__ __launch_bounds__(256) void fused_mlp_kernel(
    const float* __restrict__ Xg,   // (262144, 512)
    const float* __restrict__ W1,   // (512, 512)
    const float* __restrict__ b1,   // (512)
    const float* __restrict__ W2,   // (512, 2)
    const float* __restrict__ b2,   // (2)
    float* __restrict__ outD,       // d_out + 0       : D_init (512x512)
    float* __restrict__ outW,       // d_out + 262144  : W      (512x512)
    float* __restrict__ embPartial) // ws: one partial per block (4096)
{
  __shared__ float smem[MB * SA + NSIDE * SB]; // 11520 floats (~45KB)
  __shared__ float red[64];
  float* ldsA = smem;            // [64][20]  (K-contiguous)
  float* ldsB = smem + MB * SA;  // [512][20] (transposed W1 chunk, K-contiguous)
  float* ldsH = smem;            // [64][65]  reused after the K loop

  const int t    = threadIdx.x;
  const int lane = t & 31;
  const int wave = t >> 5;
  const int l16  = lane & 15;
  const int lh   = lane >> 4;    // 0: K+{0,1}  1: K+{2,3}   (ISA f32 A/B layout)
  const int mt   = wave & 3;     // m-tile 0..3 (16 rows each)
  const int ch   = wave >> 2;    // column half of each 64-wide n-block
  const int rbase = blockIdx.x * MB;

  v8f acc[8][2];
#pragma unroll
  for (int nb = 0; nb < 8; ++nb)
#pragma unroll
    for (int p = 0; p < 2; ++p)
      acc[nb][p] = (v8f){0.f, 0.f, 0.f, 0.f, 0.f, 0.f, 0.f, 0.f};

  for (int k0 = 0; k0 < NSIDE; k0 += KC) {
    __syncthreads();
    // stage A chunk: 64 rows x 16 k  (coalesced 64B per 16 threads)
#pragma unroll
    for (int i = 0; i < 4; ++i) {
      int idx = t + i * 256;
      int r = idx >> 4, kk = idx & 15;
      ldsA[r * SA + kk] = Xg[(rbase + r) * NSIDE + k0 + kk];
    }
    // stage B chunk transposed: ldsB[col][kk] = W1[k0+kk][col]
#pragma unroll
    for (int i = 0; i < 32; ++i) {
      int idx = t + i * 256;
      int kk = idx >> 9, c = idx & 511;
      ldsB[c * SB + kk] = W1[(k0 + kk) * NSIDE + c];
    }
    if (k0 + KC < NSIDE) // hint next A chunk toward the caches
      __builtin_prefetch(&Xg[(rbase + (t >> 4)) * NSIDE + (k0 + KC) + (t & 15)], 0, 1);
    __syncthreads();

#pragma unroll
    for (int kk = 0; kk < KC; kk += 4) {
      const int ko = kk + lh * 2;
      v2f a = *(const v2f*)&ldsA[(mt * 16 + l16) * SA + ko];
#pragma unroll
      for (int nb = 0; nb < 8; ++nb) {
#pragma unroll
        for (int p = 0; p < 2; ++p) {
          int col = nb * 64 + ch * 32 + p * 16 + l16;
          v2f b = *(const v2f*)&ldsB[col * SB + ko];
          acc[nb][p] = __builtin_amdgcn_wmma_f32_16x16x4_f32(
              false, a, false, b, (short)0, acc[nb][p], false, false);
        }
      }
    }
  }

  // ---- epilogue: relu + bias -> LDS tile; per-row dots with W2; emb partial
  float eacc = 0.f;  // threads 0..127: row t&63, output col j = t>>6
  float hsum = 0.f;  // threads 128..191: row (t-128) running sum of h
  for (int nb = 0; nb < 8; ++nb) {
    __syncthreads();
#pragma unroll
    for (int p = 0; p < 2; ++p) {
      int cloc = ch * 32 + p * 16 + l16;      // 0..63 inside the n-block
      float bias = b1[nb * 64 + cloc];
#pragma unroll
      for (int v = 0; v < 8; ++v) {
        int rloc = mt * 16 + lh * 8 + v;      // C/D VGPR layout (ISA 7.12.2)
        float hv = acc[nb][p][v] + bias;
        ldsH[rloc * SH + cloc] = hv > 0.f ? hv : 0.f;
      }
    }
    __syncthreads();
    if (t < 128) {
      int r = t & 63, j = t >> 6;
      float s = 0.f;
#pragma unroll 4
      for (int c = 0; c < 64; ++c)
        s += ldsH[r * SH + c] * W2[(nb * 64 + c) * 2 + j];
      eacc += s;
    } else if (t < 192) {
      int r = t - 128;
      float s = 0.f;
#pragma unroll 4
      for (int c = 0; c < 64; ++c) s += ldsH[r * SH + c];
      hsum += s;
    }
  }
  __syncthreads();
  if (t >= 128 && t < 192) red[t - 128] = hsum;
  __syncthreads();
  if (t == 0) {
    float s = 0.f;
    for (int i = 0; i < 64; ++i) s += red[i];   // fixed order -> deterministic
    embPartial[blockIdx.x] = s;
  }
  if (t < 128) {
    int r = t & 63, j = t >> 6;
    int R = rbase + r;
    float x = eacc + b2[j];
    float sp = fmaxf(x, 0.f) + log1pf(expf(-fabsf(x)));  // stable softplus
    if (j == 0) {
      int ii = R >> 9, jj = R & 511;
      outD[R] = (ii == jj) ? 0.f : sp;   // zero diagonal
    } else {
      outW[R] = sp;
    }
  }
}

// ---------------------------------------------------------------------------
// Kernel 2: per-index row mean (D_col) and column mean (D_row) of D_init
// ---------------------------------------------------------------------------
__global__ __launch_bounds__(64) void rowcol_kernel(
    const float* __restrict__ D, float* __restrict__ D_row, float* __restrict__ D_col)
{
  __shared__ float r1[64], r2[64];
  const int i = blockIdx.x, t = threadIdx.x;
  float sr = 0.f, sc = 0.f;
  for (int j = t; j < NSIDE; j += 64) {
    sr += D[i * NSIDE + j];   // row i sum    -> D_col[i] (mean over axis=1)
    sc += D[j * NSIDE + i];   // column i sum -> D_row[i] (mean over axis=0)
  }
  r1[t] = sr; r2[t] = sc;
  __syncthreads();
  for (int off = 32; off > 0; off >>= 1) {
    if (t < off) { r1[t] += r1[t + off]; r2[t] += r2[t + off]; }
    __syncthreads();
  }
  if (t == 0) {
    D_col[i] = r1[0] * (1.f / 512.f);
    D_row[i] = r2[0] * (1.f / 512.f);
  }
}

// ---------------------------------------------------------------------------
// Kernel 3: emb = sum(embPartial);  D_mean = mean(D_col)
// ---------------------------------------------------------------------------
__global__ __launch_bounds__(256) void finalize_scalars_kernel(
    const float* __restrict__ embPartial, const float* __restrict__ D_col,
    float* __restrict__ emb_out, float* __restrict__ D_mean)
{
  __shared__ float red[256];
  const int t = threadIdx.x;
  float s = 0.f;
  for (int i = t; i < 4096; i += 256) s += embPartial[i];
  red[t] = s;
  __syncthreads();
  for (int off = 128; off > 0; off >>= 1) {
    if (t < off) red[t] += red[t + off];
    __syncthreads();
  }
  if (t == 0) emb_out[0] = red[0];
  __syncthreads();
  red[t] = D_col[t] + D_col[t + 256];
  __syncthreads();
  for (int off = 128; off > 0; off >>= 1) {
    if (t < off) red[t] += red[t + off];
    __syncthreads();
  }
  if (t == 0) D_mean[0] = red[0] * (1.f / 512.f);
}

// ---------------------------------------------------------------------------
// Kernel 4: A_lr = B = -0.5*(D - D_row - D_col + D_mean)
// ---------------------------------------------------------------------------
__global__ __launch_bounds__(256) void buildB_kernel(
    const float* __restrict__ D, const float* __restrict__ D_row,
    const float* __restrict__ D_col, const float* __restrict__ D_mean,
    float* __restrict__ A_lr)
{
  const int idx = blockIdx.x * 256 + threadIdx.x;   // grid covers 262144
  const int i = idx >> 9, j = idx & 511;
  A_lr[idx] = -0.5f * (D[idx] - D_row[j] - D_col[i] + D_mean[0]);
}

// ---------------------------------------------------------------------------
// Kernel 5: power iteration + deflation (single workgroup, fully sequential
// semantics matching the reference; fixed-order reductions -> deterministic)
// ---------------------------------------------------------------------------
__global__ __launch_bounds__(256) void power_iter_kernel(
    float* __restrict__ A_lr,          // ws: 512x512, pre-built B (mutated)
    const float* __restrict__ u_init,  // (512, 3)
    const float* __restrict__ noise_x, // (512, 3)
    float* __restrict__ X0)            // ws: initial X (512x3)
{
  __shared__ float uc[512];   // current u
  __shared__ float un[512];   // normalized u
  __shared__ float red[256];
  const int t = threadIdx.x;

  for (int k = 0; k < 3; ++k) {
    for (int i = t; i < 512; i += 256) uc[i] = u_init[i * 3 + k];
    __syncthreads();
    for (int s = 0; s < 10; ++s) {
      // norm = ||u||
      red[t] = uc[t] * uc[t] + uc[t + 256] * uc[t + 256];
      __syncthreads();
      for (int off = 128; off > 0; off >>= 1) {
        if (t < off) red[t] += red[t + off];
        __syncthreads();
      }
      const float d = fmaxf(sqrtf(red[0]), 0.001f);
      un[t] = uc[t] / d;
      un[t + 256] = uc[t + 256] / d;
      __syncthreads();
      // u = A_lr @ un  (each thread: 2 rows)
      float a0 = 0.f, a1 = 0.f;
      for (int j = 0; j < 512; ++j) {
        const float uj = un[j];
        a0 += A_lr[t * 512 + j] * uj;
        a1 += A_lr[(t + 256) * 512 + j] * uj;
      }
      uc[t] = a0;
      uc[t + 256] = a1;
      __syncthreads();
    }
    // scale by (eig_sq + 0.01)^0.25
    red[t] = uc[t] * uc[t] + uc[t + 256] * uc[t + 256];
    __syncthreads();
    for (int off = 128; off > 0; off >>= 1) {
      if (t < off) red[t] += red[t + off];
      __syncthreads();
    }
    const float fac = powf(red[0] + 0.01f, 0.25f);
    uc[t] *= fac;
    uc[t + 256] *= fac;
    __syncthreads();
    // X column k = u + noise
    for (int i = t; i < 512; i += 256) X0[i * 3 + k] = uc[i] + noise_x[i * 3 + k];
    // deflate: A_lr -= u u^T
    for (int e = t; e < NROWS; e += 256) {
      const int r = e >> 9, c = e & 511;
      A_lr[e] -= uc[r] * uc[c];
    }
    __syncthreads();
  }
}

// ---------------------------------------------------------------------------
// Kernel 6: one gradient-descent layout step (one block per row i)
// ---------------------------------------------------------------------------
__global__ __launch_bounds__(128) void layout_step_kernel(
    const float* __restrict__ Xsrc, float* __restrict__ Xdst,
    const float* __restrict__ D, const float* __restrict__ Wm, float tstep)
{
  __shared__ float red[3 * 128];
  const int i = blockIdx.x, t = threadIdx.x;
  const float xi0 = Xsrc[i * 3 + 0], xi1 = Xsrc[i * 3 + 1], xi2 = Xsrc[i * 3 + 2];
  float g0 = 0.f, g1 = 0.f, g2 = 0.f;
  for (int j = t; j < NSIDE; j += 128) {
    const float d0 = xi0 - Xsrc[j * 3 + 0];
    const float d1 = xi1 - Xsrc[j * 3 + 1];
    const float d2 = xi2 - Xsrc[j * 3 + 2];
    const float dx = sqrtf(d0 * d0 + d1 * d1 + d2 * d2 + 0.01f);
    const float wij = Wm[i * NSIDE + j], wji = Wm[j * NSIDE + i];
    const float dij = D[i * NSIDE + j], dji = D[j * NSIDE + i];
    const float coef = -2.f * (wij * (dij - dx) + wji * (dji - dx)) / dx;
    g0 += coef * d0; g1 += coef * d1; g2 += coef * d2;
  }
  red[t] = g0; red[128 + t] = g1; red[256 + t] = g2;
  __syncthreads();
  for (int off = 64; off > 0; off >>= 1) {
    if (t < off) {
      red[t] += red[t + off];
      red[128 + t] += red[128 + t + off];
      red[256 + t] += red[256 + t + off];
    }
    __syncthreads();
  }
  if (t == 0) {
    const float dX0 = -0.1f * red[0], dX1 = -0.1f * red[128], dX2 = -0.1f * red[256];
    const float speed = sqrtf(dX0 * dX0 + dX1 * dX1 + dX2 * dX2 + 0.001f);
    const float alpha = 0.1f + (5.0f - 0.1f) * ((float)T_STEPS - tstep) / (float)T_STEPS;
    const float scale = alpha * tanhf(speed / alpha) / speed;
    Xdst[i * 3 + 0] = xi0 + dX0 * scale;
    Xdst[i * 3 + 1] = xi1 + dX1 * scale;
    Xdst[i * 3 + 2] = xi2 + dX2 * scale;
  }
}

// ---------------------------------------------------------------------------
// Kernel 7: copy final X into d_out
// ---------------------------------------------------------------------------
__global__ void copyX_kernel(const float* __restrict__ Xf, float* __restrict__ out)
{
  const int i = blockIdx.x * 256 + threadIdx.x;
  if (i < NSIDE * 3) out[i] = Xf[i];
}

// ---------------------------------------------------------------------------
// Host driver (graph-capture safe: only kernel launches on `stream`)
// ---------------------------------------------------------------------------
extern "C" void kernel_launch(void* const* d_in, const int* in_sizes, int n_in,
                              void* d_out, int out_size, void* d_ws, size_t ws_size,
                              hipStream_t stream) {
  const float* gnn     = (const float*)d_in[0];  // (262144, 512)
  const float* W1      = (const float*)d_in[1];  // (512, 512)
  const float* b1      = (const float*)d_in[2];  // (512)
  const float* W2      = (const float*)d_in[3];  // (512, 2)
  const float* b2      = (const float*)d_in[4];  // (2)
  const float* noise_x = (const float*)d_in[5];  // (512, 3)
  const float* u_init  = (const float*)d_in[6];  // (512, 3)

  float* out   = (float*)d_out;
  float* outD  = out;                 // 262144
  float* outW  = out + NROWS;         // 262144
  float* emb   = out + 2 * NROWS;     // 1
  float* outX  = out + 2 * NROWS + 1; // 1536

  float* ws     = (float*)d_ws;
  float* embP   = ws;            // 4096
  float* D_row  = ws + 4096;     // 512
  float* D_col  = ws + 4608;     // 512
  float* D_mean = ws + 5120;     // 1
  float* Xb0    = ws + 6144;     // 1536
  float* Xb1    = ws + 8192;     // 1536
  float* A_lr   = ws + 16384;    // 262144  (~1.1 MB total ws usage)

  fused_mlp_kernel<<<NROWS / MB, 256, 0, stream>>>(gnn, W1, b1, W2, b2, outD, outW, embP);
  rowcol_kernel<<<NSIDE, 64, 0, stream>>>(outD, D_row, D_col);
  finalize_scalars_kernel<<<1, 256, 0, stream>>>(embP, D_col, emb, D_mean);
  buildB_kernel<<<NROWS / 256, 256, 0, stream>>>(outD, D_row, D_col, D_mean, A_lr);
  power_iter_kernel<<<1, 256, 0, stream>>>(A_lr, u_init, noise_x, Xb0);

  float* src = Xb0;
  float* dst = Xb1;
  for (int t = 0; t < T_STEPS; ++t) {
    layout_step_kernel<<<NSIDE, 128, 0, stream>>>(src, dst, outD, outW, (float)t);
    float* tmp = src; src = dst; dst = tmp;
  }
  copyX_kernel<<<6, 256, 0, stream>>>(src, outX);
}